// BlazeFace_1262720385666
// MI455X (gfx1250) — compile-verified
//
#include <hip/hip_runtime.h>
#include <stdint.h>

#define NUM_ANCHORS 896
#define MAX_DET     64
#define BSCALE      128.0f
#define MIN_SCORE   0.5f
#define IOU_THRESH  0.3f
#define BLOCK       256
#define NWAVES      (BLOCK / 32)
#define TILE_ELEMS  (NUM_ANCHORS * 16)            // 14336 f32 = 57344 B

typedef unsigned int tdm_v4u __attribute__((ext_vector_type(4)));
typedef int          tdm_v8i __attribute__((ext_vector_type(8)));
typedef int          tdm_v4i __attribute__((ext_vector_type(4)));

__global__ __launch_bounds__(BLOCK) void blazeface_nms_kernel(
    const float* __restrict__ raw_boxes,   // (B, 896, 16)
    const float* __restrict__ raw_scores,  // (B, 896)
    const float* __restrict__ anchors,     // (896, 4)
    const float* __restrict__ tmatrix,     // (B, 8)
    const int*   __restrict__ hptr,        // scalar
    const int*   __restrict__ wptr,        // scalar
    float* __restrict__ out)               // (B, 64, 17)
{
    __shared__ float sCoords[TILE_ELEMS];         // raw tile, decoded in place (57344 B)
    __shared__ float sRawScore[NUM_ANCHORS];      // 3584 B
    __shared__ float sRedV[NWAVES];
    __shared__ int   sRedI[NWAVES];
    __shared__ float sPart[NWAVES][17];

    const int tid  = threadIdx.x;
    const int lane = tid & 31;
    const int wave = tid >> 5;
    const int b    = blockIdx.x;

    const float* gsrc = raw_boxes + (size_t)b * TILE_ELEMS;

    // ---- 1) Bulk tile via Tensor Data Mover (TENSORcnt path); fallback: async-to-LDS loop
#if defined(__has_builtin) && __has_builtin(__builtin_amdgcn_tensor_load_to_lds)
    if (tid < 32) {                                // wave-uniform guard: one TDM op total
        uint32_t ldsAddr = (uint32_t)(uintptr_t)(&sCoords[0]);
        uint64_t ga      = (uint64_t)(uintptr_t)gsrc;
        tdm_v4u g0;
        g0.x = 1u;                                             // count=1, user descriptor
        g0.y = ldsAddr;                                        // lds_addr (bytes)
        g0.z = (uint32_t)(ga & 0xFFFFFFFFu);                   // global_addr[31:0]
        g0.w = (uint32_t)((ga >> 32) & 0x01FFFFFFu)            // global_addr[56:32]
             | 0x80000000u;                                    // type=2 ("image") in [127:126]
        tdm_v8i g1;
        g1[0] = 0x00020000;                                    // data_size=2 (4 bytes), no mask
        g1[1] = (int)((uint32_t)TILE_ELEMS << 16);             // tensor_dim0 lo16 @ [63:48]
        g1[2] = (int)(1u << 16);                               // dim0 hi16=0; tensor_dim1=1 lo16 @ [95:80]
        g1[3] = (int)((uint32_t)TILE_ELEMS << 16);             // dim1 hi16=0; tile_dim0 @ [127:112]
        g1[4] = 1;                                             // tile_dim1=1, tile_dim2=0
        g1[5] = TILE_ELEMS;                                    // tensor_dim0_stride lo32
        g1[6] = (int)((uint32_t)TILE_ELEMS << 16);             // stride0 hi16=0; stride1 lo16
        g1[7] = 0;                                             // stride1 hi32
        tdm_v4i gz4 = {0, 0, 0, 0};
        tdm_v8i gz8 = {0, 0, 0, 0, 0, 0, 0, 0};
        // 6-arg form (clang-23 / therock-10.0 headers): groups 2/3 disabled (zeros)
        __builtin_amdgcn_tensor_load_to_lds(g0, g1, gz4, gz4, gz8, 0);
    }
#else
    for (int c = tid; c < TILE_ELEMS / 4; c += BLOCK) {
        uint32_t lds_off = (uint32_t)(uintptr_t)(&sCoords[c * 4]);
        uint64_t gaddr   = (uint64_t)(uintptr_t)(gsrc + c * 4);
        asm volatile("global_load_async_to_lds_b128 %0, %1, off"
                     :: "v"(lds_off), "v"(gaddr) : "memory");
    }
#endif

    // ---- 2) Scores row via gfx1250 async-to-LDS (ASYNCcnt path)
    for (int c = tid; c < NUM_ANCHORS / 4; c += BLOCK) {
        uint32_t lds_off = (uint32_t)(uintptr_t)(&sRawScore[c * 4]);
        uint64_t gaddr   = (uint64_t)(uintptr_t)(raw_scores + (size_t)b * NUM_ANCHORS + c * 4);
        asm volatile("global_load_async_to_lds_b128 %0, %1, off"
                     :: "v"(lds_off), "v"(gaddr) : "memory");
    }

    asm volatile("s_wait_asynccnt 0" ::: "memory");
#if defined(__has_builtin) && __has_builtin(__builtin_amdgcn_tensor_load_to_lds)
    if (tid < 32) __builtin_amdgcn_s_wait_tensorcnt(0);
#endif
    __syncthreads();

    // ---- 3) Per-thread anchor ownership: a = tid + j*256, j < nOwn (896 = 3*256 + 128)
    const int nOwn = (tid < (NUM_ANCHORS - 3 * BLOCK)) ? 4 : 3;
    float rRem[4], rSc[4];
    float rby0[4], rbx0[4], rby1[4], rbx1[4];
    #pragma unroll
    for (int j = 0; j < 4; ++j) { rRem[j] = -1e30f; rSc[j] = 0.0f;
                                  rby0[j] = rbx0[j] = rby1[j] = rbx1[j] = 0.0f; }

    const float inv = 1.0f / BSCALE;
    for (int j = 0; j < nOwn; ++j) {
        const int a = tid + j * BLOCK;
        // sigmoid score (registers only)
        float s = sRawScore[a];
        s = fminf(fmaxf(s, -100.0f), 100.0f);
        s = 1.0f / (1.0f + __expf(-s));
        rSc[j]  = s;
        rRem[j] = (s >= MIN_SCORE) ? s : -1.0f;
        // decode in place in LDS, keep box corners in registers
        float ax = anchors[a * 4 + 0];
        float ay = anchors[a * 4 + 1];
        float aw = anchors[a * 4 + 2];
        float ah = anchors[a * 4 + 3];
        float* r = &sCoords[a * 16];
        float rb[16];
        #pragma unroll
        for (int k = 0; k < 16; ++k) rb[k] = r[k];
        float xc = rb[0] * inv * aw + ax;
        float yc = rb[1] * inv * ah + ay;
        float bw = rb[2] * inv * aw;
        float bh = rb[3] * inv * ah;
        rby0[j] = yc - 0.5f * bh;  rbx0[j] = xc - 0.5f * bw;
        rby1[j] = yc + 0.5f * bh;  rbx1[j] = xc + 0.5f * bw;
        r[0] = rby0[j]; r[1] = rbx0[j]; r[2] = rby1[j]; r[3] = rbx1[j];
        #pragma unroll
        for (int k = 0; k < 6; ++k) {
            r[4 + 2 * k] = rb[4 + 2 * k] * inv * aw + ax;   // kp x
            r[5 + 2 * k] = rb[5 + 2 * k] * inv * ah + ay;   // kp y
        }
    }
    __syncthreads();

    const float hf = (float)hptr[0];
    const float wf = (float)wptr[0];
    float* outb = out + (size_t)b * MAX_DET * 17;

    // ---- 4) 64 sequential weighted-NMS steps; 2 barriers per step
    for (int d = 0; d < MAX_DET; ++d) {
        // 4a) argmax of rem: registers -> wave32 shuffle -> per-wave LDS partial
        float bv = rRem[0]; int bi = tid;
        #pragma unroll
        for (int j = 1; j < 4; ++j)
            if (rRem[j] > bv) { bv = rRem[j]; bi = tid + j * BLOCK; }
        #pragma unroll
        for (int off = 16; off > 0; off >>= 1) {
            float ov = __shfl_down(bv, off, 32);
            int   oi = __shfl_down(bi, off, 32);
            if (ov > bv || (ov == bv && oi < bi)) { bv = ov; bi = oi; }
        }
        if (lane == 0) { sRedV[wave] = bv; sRedI[wave] = bi; }
        __syncthreads();                                    // barrier A

        // every thread combines the 8 wave partials (cheap, removes a barrier)
        float fbv = sRedV[0]; int fbi = sRedI[0];
        #pragma unroll
        for (int i = 1; i < NWAVES; ++i)
            if (sRedV[i] > fbv || (sRedV[i] == fbv && sRedI[i] < fbi)) { fbv = sRedV[i]; fbi = sRedI[i]; }
        const float by0 = sCoords[fbi * 16 + 0];            // broadcast LDS reads
        const float bx0 = sCoords[fbi * 16 + 1];
        const float by1 = sCoords[fbi * 16 + 2];
        const float bx1 = sCoords[fbi * 16 + 3];
        const float best  = fbv;
        const float area1 = fmaxf(by1 - by0, 0.0f) * fmaxf(bx1 - bx0, 0.0f);

        // 4b) IoU mask + weighted partial sums over owned anchors (boxes in registers)
        float acc[17];
        #pragma unroll
        for (int k = 0; k < 17; ++k) acc[k] = 0.0f;         // acc[16] = denom
        for (int j = 0; j < nOwn; ++j) {
            float iy0 = fmaxf(by0, rby0[j]), ix0 = fmaxf(bx0, rbx0[j]);
            float iy1 = fminf(by1, rby1[j]), ix1 = fminf(bx1, rbx1[j]);
            float inter = fmaxf(iy1 - iy0, 0.0f) * fmaxf(ix1 - ix0, 0.0f);
            float area2 = fmaxf(rby1[j] - rby0[j], 0.0f) * fmaxf(rbx1[j] - rbx0[j], 0.0f);
            float iou = inter / fmaxf(area1 + area2 - inter, 1e-6f);
            if (iou > IOU_THRESH && rRem[j] > 0.0f) {
                const int a = tid + j * BLOCK;
                float wt = rSc[j];
                acc[16] += wt;
                const float* c = &sCoords[a * 16];
                #pragma unroll
                for (int k = 0; k < 16; ++k) acc[k] += wt * c[k];
                rRem[j] = -1.0f;                            // suppress (register-private)
            }
        }
        #pragma unroll
        for (int k = 0; k < 17; ++k) {
            float v = acc[k];
            #pragma unroll
            for (int off = 16; off > 0; off >>= 1) v += __shfl_down(v, off, 32);
            if (lane == 0) sPart[wave][k] = v;
        }
        __syncthreads();                                    // barrier B

        // 4c) finalize: blend, validity, projection, rescale, write 17 floats
        if (tid == 0) {
            float tot[17];
            #pragma unroll
            for (int k = 0; k < 17; ++k) {
                float v = 0.0f;
                #pragma unroll
                for (int i = 0; i < NWAVES; ++i) v += sPart[i][k];
                tot[k] = v;
            }
            float denom = fmaxf(tot[16], 1e-6f);
            float dv[17];
            #pragma unroll
            for (int k = 0; k < 16; ++k) dv[k] = tot[k] / denom;
            dv[16] = best;
            float o[17];
            if (best > 0.0f) {
                const float* mm = tmatrix + (size_t)b * 8;
                float m0 = mm[0], m1 = mm[1], m3 = mm[3];
                float m4 = mm[4], m5 = mm[5], m7 = mm[7];
                const int xi[8] = {1, 3, 4, 6, 8, 10, 12, 14};
                const int yi[8] = {0, 2, 5, 7, 9, 11, 13, 15};
                #pragma unroll
                for (int k = 0; k < 17; ++k) o[k] = dv[k];
                #pragma unroll
                for (int p = 0; p < 8; ++p) {
                    float x = dv[xi[p]], y = dv[yi[p]];
                    o[xi[p]] = x * m0 + y * m1 + m3;
                    o[yi[p]] = x * m4 + y * m5 + m7;
                }
                // rescale: [h, w, h, w, (w,h)x6, 1]
                o[0] *= hf; o[1] *= wf; o[2] *= hf; o[3] *= wf;
                #pragma unroll
                for (int p = 0; p < 6; ++p) { o[4 + 2 * p] *= wf; o[5 + 2 * p] *= hf; }
            } else {
                #pragma unroll
                for (int k = 0; k < 17; ++k) o[k] = 0.0f;
            }
            #pragma unroll
            for (int k = 0; k < 17; ++k) outb[d * 17 + k] = o[k];
        }
        // no trailing barrier needed: next step touches sRedV/sRedI only after barrier B,
        // and rem/score/box state is register-private.
    }
}

extern "C" void kernel_launch(void* const* d_in, const int* in_sizes, int n_in,
                              void* d_out, int out_size, void* d_ws, size_t ws_size,
                              hipStream_t stream) {
    const float* raw_boxes  = (const float*)d_in[0];
    const float* raw_scores = (const float*)d_in[1];
    const float* anchors    = (const float*)d_in[2];
    const float* tmat       = (const float*)d_in[3];
    const int*   hptr       = (const int*)d_in[4];
    const int*   wptr       = (const int*)d_in[5];

    const int B = in_sizes[0] / (NUM_ANCHORS * 16);

    blazeface_nms_kernel<<<B, BLOCK, 0, stream>>>(
        raw_boxes, raw_scores, anchors, tmat, hptr, wptr, (float*)d_out);
}